// PartCEMV1_7894149890693
// MI455X (gfx1250) — compile-verified
//
#include <hip/hip_runtime.h>
#include <hip/hip_bf16.h>
#include <math.h>

typedef __attribute__((ext_vector_type(2))) float v2f;
typedef __attribute__((ext_vector_type(8))) float v8f;
typedef __attribute__((ext_vector_type(4))) unsigned int v4u;
typedef __attribute__((ext_vector_type(8))) int v8i;
typedef __attribute__((ext_vector_type(4))) int v4i;

#define NB   128   // batch
#define NC   2048  // channels
#define NN   196   // spatial (14*14)
#define NK   112   // concepts
#define NKP1 113   // concepts + background
#define NCOL 128   // padded GEMM columns (0..111 concepts, 112 bg, 113 w_cfc, rest 0)
#define NCLS 200
#define MTOT (NB * NN)       // 25088
#define MBLK 64
#define NBLK_M (MTOT / MBLK) // 392

// LDS strides (in v2f units) chosen so lane-halves (K0/K1 vs K2/K3) use disjoint banks:
// stride in dwords must be ≡ 32 (mod 64).  A: 160 dw = 80 v2f.  B: 288 dw = 144 v2f.
#define A2_STRIDE 80
#define BL_STRIDE 144

// ---------------------------------------------------------------- TDM issue helper
// Loads one contiguous 4096-float (16 KB) B tile into LDS, inserting 32 dwords of
// padding after every 256 dwords (pad_interval code 7 = 256 dw, pad_amount code 31 = 32 dw),
// which produces the 288-dword per-K-pair pitch of Bl.
static __device__ inline void tdm_load_b_tile(const float* __restrict__ src, unsigned lds_off) {
    unsigned long long ga = (unsigned long long)(uintptr_t)src;
    v4u g0;
    g0.x = 1u;                                            // count=1, user desc
    g0.y = lds_off;                                       // lds_addr
    g0.z = (unsigned int)ga;                              // global_addr[31:0]
    g0.w = (unsigned int)((ga >> 32) & 0x01ffffffu)       // global_addr[56:32]
         | 0x80000000u;                                   // type=2 ("image")
    v8i g1;
    g1[0] = (int)((2u << 16)      // data_size = 4B
                | (1u << 20)      // pad_enable
                | (7u << 22)      // pad_interval: 256 dwords
                | (31u << 25));   // pad_amount: 32 dwords
    g1[1] = (int)(4096u << 16);   // tensor_dim0 = 4096 (low 16 bits at [31:16])
    g1[2] = (int)(1u << 16);      // tensor_dim0 hi = 0 | tensor_dim1 = 1
    g1[3] = (int)(4096u << 16);   // tensor_dim1 hi = 0 | tile_dim0 = 4096
    g1[4] = 1;                    // tile_dim1 = 1, tile_dim2 = 0 (unused)
    g1[5] = 4096;                 // tensor_dim0_stride low32
    g1[6] = 0;
    g1[7] = 0;
    v4i z4 = {0, 0, 0, 0};
#if __has_include(<hip/amd_detail/amd_gfx1250_TDM.h>)
    v8i z8 = {0, 0, 0, 0, 0, 0, 0, 0};
    __builtin_amdgcn_tensor_load_to_lds(g0, g1, z4, z4, z8, 0);
#else
    __builtin_amdgcn_tensor_load_to_lds(g0, g1, z4, z4, 0);
#endif
}

// ---------------------------------------------------------------- kernel: build BmTP (pair-interleaved)
// BmTP flat index i = p*256 + j*2 + v  holds B[channel c=2p+v][concept col j]
__global__ __launch_bounds__(256) void build_bmat(const float* __restrict__ concepts,
                                                  const float* __restrict__ background,
                                                  const float* __restrict__ w_cfc,
                                                  float* __restrict__ BmTP) {
    int i = blockIdx.x * 256 + threadIdx.x;
    if (i >= NC * NCOL) return;
    int p   = i >> 8;
    int rem = i & 255;
    int j   = rem >> 1;
    int c   = (p << 1) | (rem & 1);
    float v = 0.0f;
    if (j < NK)           v = concepts[(size_t)j * NC + c];
    else if (j == NK)     v = background[c];
    else if (j == NK + 1) v = w_cfc[c];
    BmTP[i] = v;
}

// ---------------------------------------------------------------- kernel: a_sq[113]
__global__ __launch_bounds__(256) void asq_kernel(const float* __restrict__ concepts,
                                                  const float* __restrict__ background,
                                                  float* __restrict__ asq) {
    int k = blockIdx.x;
    const float* row = (k < NK) ? (concepts + (size_t)k * NC) : background;
    float s = 0.0f;
    for (int c = threadIdx.x; c < NC; c += 256) { float v = row[c]; s += v * v; }
    __shared__ float sm[256];
    sm[threadIdx.x] = s;
    __syncthreads();
    for (int off = 128; off > 0; off >>= 1) {
        if ((int)threadIdx.x < off) sm[threadIdx.x] += sm[threadIdx.x + off];
        __syncthreads();
    }
    if (threadIdx.x == 0) asq[k] = sm[0];
}

// ---------------------------------------------------------------- kernel: fused GEMM + dist + softmax
__global__ __launch_bounds__(256) void gemm_softmax(const float* __restrict__ x,
                                                    const float* __restrict__ BmTP,
                                                    const float* __restrict__ asq,
                                                    float* __restrict__ out1,
                                                    float* __restrict__ part) {
    __shared__ v2f   A2[16 * A2_STRIDE];   // K-pair-interleaved A tile: 10240 B
    __shared__ v2f   Bl[16 * BL_STRIDE];   // TDM-filled, padded B tile:  18432 B
    __shared__ float Dlds[64][129];        // result tile: 33024 B
    __shared__ float xsqp[256];
    __shared__ float asql[128];
    __shared__ float rxw[64];

    const int t  = threadIdx.x;
    const int bi = blockIdx.x;
    const int m0 = bi * MBLK;

    // ---- A loader roles: row ra, channel-quarter qa (8 channels = 4 pairs each)
    const int ra = t & 63;
    const int qa = t >> 6;

    const int mrow = m0 + ra;
    const int brow = mrow / NN;
    const int nrow = mrow - brow * NN;
    const float* __restrict__ arow = x + (size_t)brow * ((size_t)NC * NN) + nrow;

    if (t < NKP1) asql[t] = asq[t];

    // ---- wave tiling: wave w -> row-tile (w>>1)*16, col-group (w&1)*64 (4 tiles of 16)
    const int wave = t >> 5;
    const int lane = t & 31;
    const int lrow = lane & 15;
    const int lhi  = lane >> 4;
    const int r0   = (wave >> 1) * 16;
    const int cg   = (wave & 1) * 64;

    const unsigned bl_off = (unsigned)(uintptr_t)(void*)Bl;   // LDS byte offset of Bl

    v8f acc[4] = {};
    float sumsq = 0.0f;

    for (int k0 = 0; k0 < NC; k0 += 32) {
        // async B-tile DMA: pairs k0/2 .. k0/2+15, contiguous 4096 floats in BmTP
        if (wave == 0)
            tdm_load_b_tile(BmTP + (size_t)k0 * NCOL, bl_off);

        // A tile: 64 rows x 32 channels, stored as v2f channel-pairs
#pragma unroll
        for (int jj = 0; jj < 4; ++jj) {
            const int c0 = qa * 8 + jj * 2;
            const float v0 = arow[(size_t)(k0 + c0) * NN];
            const float v1 = arow[(size_t)(k0 + c0 + 1) * NN];
            v2f pv; pv.x = v0; pv.y = v1;
            A2[(qa * 4 + jj) * A2_STRIDE + ra] = pv;
            sumsq += v0 * v0 + v1 * v1;
        }

        if (wave == 0) __builtin_amdgcn_s_wait_tensorcnt(0);
        __syncthreads();

#pragma unroll
        for (int kk = 0; kk < 8; ++kk) {
            const int p = kk * 2 + lhi;                 // K-pair index (lanes 0-15: K0/K1, 16-31: K2/K3)
            const v2f a = A2[p * A2_STRIDE + r0 + lrow];
#pragma unroll
            for (int jt = 0; jt < 4; ++jt) {
                const v2f bv = Bl[p * BL_STRIDE + cg + jt * 16 + lrow];
                acc[jt] = __builtin_amdgcn_wmma_f32_16x16x4_f32(
                    false, a, false, bv, (short)0, acc[jt], false, false);
            }
        }
        __syncthreads();
    }

    // ---- dump accumulators: VGPR v holds M = lhi*8 + v, N = lrow
#pragma unroll
    for (int jt = 0; jt < 4; ++jt)
#pragma unroll
        for (int v = 0; v < 8; ++v)
            Dlds[r0 + lhi * 8 + v][cg + jt * 16 + lrow] = acc[jt][v];
    xsqp[t] = sumsq;
    __syncthreads();

    const int bfirst = m0 / NN;
    const int split  = (bfirst + 1) * NN - m0;   // local rows [0,split) belong to bfirst

    // ---- per-row: distances, stabilized softmax over 113 concepts, write score maps
    if (t < MBLK) {
        const float xsq = xsqp[t] + xsqp[t + 64] + xsqp[t + 128] + xsqp[t + 192];
        const float xw  = Dlds[t][NKP1];         // col 113 = x . w_cfc
        float dmin = 3.402823e38f;
#pragma unroll 1
        for (int k = 0; k < NKP1; ++k) {
            const float d2 = asql[k] + xsq - 2.0f * Dlds[t][k];
            const float d  = sqrtf(fmaxf(d2, 0.0f));
            Dlds[t][k] = d;
            dmin = fminf(dmin, d);
        }
        float s = 0.0f;
#pragma unroll 1
        for (int k = 0; k < NKP1; ++k) {
            const float e = expf(dmin - Dlds[t][k]);
            Dlds[t][k] = e;                       // keep unnormalized exp
            s += e;
        }
        const float rinv = 1.0f / s;
        rxw[t] = xw * rinv;

        const int m = m0 + t;
        const int b = m / NN;
        const int n = m - b * NN;
        float* __restrict__ o1 = out1 + (size_t)b * ((size_t)NKP1 * NN) + n;
#pragma unroll 1
        for (int k = 0; k < NKP1; ++k)
            o1[(size_t)k * NN] = Dlds[t][k] * rinv;   // attn -> cpt_score_maps
    }
    __syncthreads();

    // ---- per-concept deterministic partial sums of attn*xw (for cpt_logits)
    if (t < NK) {
        float accA = 0.0f, accB = 0.0f;
#pragma unroll 1
        for (int r = 0; r < MBLK; ++r) {
            const float v = Dlds[r][t] * rxw[r];  // e * xw * rinv = attn * xw
            if (r < split) accA += v; else accB += v;
        }
        part[bi * (2 * NK) + t]      = accA;      // slot 0: image bfirst
        part[bi * (2 * NK) + NK + t] = accB;      // slot 1: image bfirst+1 (0 if unused)
    }
}

// ---------------------------------------------------------------- kernel: reduce partials + sigmoid
__global__ __launch_bounds__(256) void logits_kernel(const float* __restrict__ part,
                                                     const float* __restrict__ b_cfc,
                                                     float* __restrict__ out2) {
    int i = blockIdx.x * 256 + threadIdx.x;
    if (i >= NB * NK) return;
    int b = i / NK, k = i - b * NK;
    int lo = (b * NN) / MBLK;
    int hi = (b * NN + NN - 1) / MBLK;
    float s = 0.0f;
    for (int bi = lo; bi <= hi; ++bi) {
        int m0 = bi * MBLK;
        int bf = m0 / NN;
        int bl = (m0 + MBLK - 1) / NN;
        if (bf == b)              s += part[bi * (2 * NK) + k];
        if (bl == b && bl != bf)  s += part[bi * (2 * NK) + NK + k];
    }
    float z = s + b_cfc[0];
    out2[i] = 1.0f / (1.0f + expf(-z));
}

// ---------------------------------------------------------------- kernel: P/R precompute over w_lfc
__global__ __launch_bounds__(256) void pr_kernel(const float* __restrict__ w_lfc,
                                                 const float* __restrict__ concepts,
                                                 const float* __restrict__ modulation,
                                                 float* __restrict__ P,
                                                 float* __restrict__ R) {
    int pair = blockIdx.x;               // 0..22399 = o*112 + k
    int o = pair / NK, k = pair - o * NK;
    const float* __restrict__ wrow = w_lfc + (size_t)o * ((size_t)NK * NC) + (size_t)k * NC;
    const float* __restrict__ crow = concepts + (size_t)k * NC;
    float p = 0.0f, rr = 0.0f;
    for (int c = threadIdx.x; c < NC; c += 256) {
        float w  = wrow[c];
        float mo = modulation[c];
        p  += w * (crow[c] - mo);
        rr += w * mo;
    }
    __shared__ float sp[256], sr[256];
    sp[threadIdx.x] = p; sr[threadIdx.x] = rr;
    __syncthreads();
    for (int off = 128; off > 0; off >>= 1) {
        if ((int)threadIdx.x < off) {
            sp[threadIdx.x] += sp[threadIdx.x + off];
            sr[threadIdx.x] += sr[threadIdx.x + off];
        }
        __syncthreads();
    }
    if (threadIdx.x == 0) { P[pair] = sp[0]; R[pair] = sr[0]; }
}

// ---------------------------------------------------------------- kernel: label logits
__global__ __launch_bounds__(256) void label_kernel(const float* __restrict__ P,
                                                    const float* __restrict__ R,
                                                    const float* __restrict__ g,
                                                    const float* __restrict__ b_lfc,
                                                    float* __restrict__ out3) {
    int i = blockIdx.x * 256 + threadIdx.x;
    if (i >= NB * NCLS) return;
    int b = i / NCLS, o = i - b * NCLS;
    const float* __restrict__ Po = P + o * NK;
    const float* __restrict__ Ro = R + o * NK;
    const float* __restrict__ gb = g + b * NK;
    float acc = b_lfc[o];
    for (int k = 0; k < NK; ++k) acc += Ro[k] + gb[k] * Po[k];
    out3[i] = acc;
}

// ---------------------------------------------------------------- launch
extern "C" void kernel_launch(void* const* d_in, const int* in_sizes, int n_in,
                              void* d_out, int out_size, void* d_ws, size_t ws_size,
                              hipStream_t stream) {
    (void)in_sizes; (void)n_in; (void)out_size; (void)ws_size;
    const float* x          = (const float*)d_in[0];
    const float* concepts   = (const float*)d_in[1];
    const float* modulation = (const float*)d_in[2];
    const float* background = (const float*)d_in[3];
    const float* w_cfc      = (const float*)d_in[4];
    const float* b_cfc      = (const float*)d_in[5];
    const float* w_lfc      = (const float*)d_in[6];
    const float* b_lfc      = (const float*)d_in[7];

    float* out1 = (float*)d_out;                        // [128,113,196] score maps
    float* out2 = out1 + (size_t)NB * NKP1 * NN;        // [128,112] cpt_logits
    float* out3 = out2 + (size_t)NB * NK;               // [128,200] label_logits

    float* ws   = (float*)d_ws;
    float* part = ws;                                   // 392*224 partial logit sums
    float* asq  = part + (size_t)NBLK_M * 2 * NK;       // 113 (pad to 128)
    float* P    = asq + 128;                            // 22400
    float* R    = P + (size_t)NCLS * NK;                // 22400
    float* BmTP = R + (size_t)NCLS * NK;                // 2048*128 pair-interleaved

    build_bmat<<<(NC * NCOL + 255) / 256, 256, 0, stream>>>(concepts, background, w_cfc, BmTP);
    asq_kernel<<<NKP1, 256, 0, stream>>>(concepts, background, asq);
    gemm_softmax<<<NBLK_M, 256, 0, stream>>>(x, BmTP, asq, out1, part);
    logits_kernel<<<(NB * NK + 255) / 256, 256, 0, stream>>>(part, b_cfc, out2);
    pr_kernel<<<NCLS * NK, 256, 0, stream>>>(w_lfc, concepts, modulation, P, R);
    label_kernel<<<(NB * NCLS + 255) / 256, 256, 0, stream>>>(P, R, out2, b_lfc, out3);
}